// HR2HK_30906584662161
// MI455X (gfx1250) — compile-verified
//
#include <hip/hip_runtime.h>
#include <hip/hip_bf16.h>

// ---------------------------------------------------------------------------
// HR2HK assembly for MI455X (gfx1250, wave32).
//   K=16 kpoints, E=2048 edges, N=128 atoms, norb=13 -> Hk[16,1664,1664] c64.
//   Memory-bound: ~354MB output write dominates (~15us at 23.3 TB/s); phases
//   computed with V_WMMA_F32_16X16X4_F32 (kpoints[16x3] @ cell_shift[Ex3]^T,
//   K-dim padded 3->4), then hardware V_SIN_F32/V_COS_F32 on revolution-unit
//   arguments (no libm Payne-Hanek).
// ---------------------------------------------------------------------------

typedef __attribute__((ext_vector_type(2))) float v2f;
typedef __attribute__((ext_vector_type(8))) float v8f;

#define NORB 13
#define NORB2 (NORB * NORB)

// orbital -> angular-momentum block (sizes 1,1,3,3,5 ; offsets 0,1,2,5,8)
__device__ __forceinline__ int blk_of(int o) {
    if (o < 1) return 0;
    if (o < 2) return 1;
    if (o < 5) return 2;
    if (o < 8) return 3;
    return 4;
}

// F factor: 0.5 on diagonal orbital blocks, 1.0 above, 0.0 below
__device__ __forceinline__ float Ffac(int a, int b) {
    int ba = blk_of(a), bb = blk_of(b);
    return (ba < bb) ? 1.0f : ((ba == bb) ? 0.5f : 0.0f);
}

// phase = exp(-i*2*pi*d); d in revolutions. Fold to [-0.5,0.5], then HW
// transcendentals: v_cos_f32(t)=cos(2*pi*t), v_sin_f32(t)=sin(2*pi*t).
__device__ __forceinline__ void phase_of(float d, float& re, float& im) {
    float t = d - __builtin_rintf(d);
    re = __builtin_amdgcn_cosf(t);
    im = __builtin_amdgcn_sinf(-t);
}

// ---------------------------------------------------------------------------
// 1) zero the whole complex output (float4 stores)
// ---------------------------------------------------------------------------
__global__ void zero_kernel(float* __restrict__ out, long long n) {
    long long i = (((long long)blockIdx.x * blockDim.x) + threadIdx.x) * 4;
    if (i + 3 < n) {
        float4 z; z.x = 0.f; z.y = 0.f; z.z = 0.f; z.w = 0.f;
        *reinterpret_cast<float4*>(out + i) = z;
    } else {
        for (long long j = i; j < n; ++j) out[j] = 0.0f;
    }
}

// ---------------------------------------------------------------------------
// 2) phase[16,E] via V_WMMA_F32_16X16X4_F32 (specialized K==16: branch-free).
//    One wave (32 lanes) per 16-edge tile. d = kpoints[m,:] . cshift[e,:]
//    A (16x4, K padded): lanes 0-15 hold K=0,1 ; lanes 16-31 hold K=2,3(=0)
//    B (4x16):           lanes 0-15 hold K=0,1 ; lanes 16-31 hold K=2,3(=0)
//    D (16x16 f32): VGPR r, lanes 0-15 -> M=r ; lanes 16-31 -> M=r+8
// ---------------------------------------------------------------------------
__global__ void phase_wmma_kernel(const float* __restrict__ kp,
                                  const float* __restrict__ cs,
                                  float* __restrict__ phRe,
                                  float* __restrict__ phIm,
                                  int E) {
    const int lane = threadIdx.x & 31;
    const int hi   = lane >> 4;   // 0: K=0,1 ; 1: K=2,3
    const int m    = lane & 15;   // A row (kpoint) / B column (edge-in-tile)
    const int e    = blockIdx.x * 16 + m;

    v2f a, b;
    a.x = hi ? kp[m * 3 + 2] : kp[m * 3 + 0];
    a.y = hi ? 0.0f          : kp[m * 3 + 1];
    b.x = hi ? cs[e * 3 + 2] : cs[e * 3 + 0];
    b.y = hi ? 0.0f          : cs[e * 3 + 1];

    v8f acc = {};
    // 8 args: (neg_a, A, neg_b, B, c_mod, C, reuse_a, reuse_b)
    v8f d = __builtin_amdgcn_wmma_f32_16x16x4_f32(
        false, a, false, b, (short)0, acc, false, false);

    // lane holds rows M = hi*8 + r, column e; stores at stride E (K==16)
    long long base = (long long)(hi * 8) * E + e;
#pragma unroll
    for (int r = 0; r < 8; ++r) {
        float re, im;
        phase_of(d[r], re, im);
        phRe[base + (long long)r * E] = re;
        phIm[base + (long long)r * E] = im;
    }
}

// Generic fallback (any K): one thread per (k,e) dot product.
__global__ void phase_scalar_kernel(const float* __restrict__ kp,
                                    const float* __restrict__ cs,
                                    float* __restrict__ phRe,
                                    float* __restrict__ phIm,
                                    int E, int K) {
    int idx = blockIdx.x * blockDim.x + threadIdx.x;
    if (idx >= K * E) return;
    int e = idx % E, k = idx / E;
    float d = kp[k * 3 + 0] * cs[e * 3 + 0]
            + kp[k * 3 + 1] * cs[e * 3 + 1]
            + kp[k * 3 + 2] * cs[e * 3 + 2];
    float re, im;
    phase_of(d, re, im);
    phRe[idx] = re;
    phIm[idx] = im;
}

// ---------------------------------------------------------------------------
// 3) onsite diagonal blocks: Hk[k, i*13+a, i*13+b].re =
//       onsite[i,a,b]*F(a,b) + onsite[i,b,a]*F(b,a)   (imag stays 0)
//    Plain stores; runs before edge atomics (stream-ordered).
// ---------------------------------------------------------------------------
__global__ void onsite_kernel(const float* __restrict__ ons,
                              float* __restrict__ out,
                              int N, int K) {
    int idx = blockIdx.x * blockDim.x + threadIdx.x;
    int total = K * N * NORB2;
    if (idx >= total) return;
    int ab  = idx % NORB2;
    int rem = idx / NORB2;
    int i   = rem % N;
    int k   = rem / N;
    int a   = ab / NORB, b = ab % NORB;

    float val = ons[(long long)i * NORB2 + a * NORB + b] * Ffac(a, b)
              + ons[(long long)i * NORB2 + b * NORB + a] * Ffac(b, a);

    long long L   = (long long)N * NORB;
    long long row = (long long)i * NORB + a;
    long long col = (long long)i * NORB + b;
    long long pos = (((long long)k * L + row) * L + col) * 2;
    out[pos] = val;           // real part; imag already zeroed
}

// ---------------------------------------------------------------------------
// 4) edge scatter: one thread per (e, a, b) with F(a,b) != 0, loop over k.
//    hf = hopping[e,a,b]*F(a,b)
//    block (ie,je): += phase[k,e]      * hf  at (ie*13+a, je*13+b)
//    block (je,ie): += conj(phase[k,e])* hf  at (je*13+b, ie*13+a)
//    Atomics resolve duplicate (ie,je) pairs / self-edges; working set ~91MB
//    fits in the 192MB L2, so atomics stay on-chip.
// ---------------------------------------------------------------------------
__global__ void edge_kernel(const float* __restrict__ hop,
                            const int* __restrict__ eidx,
                            const float* __restrict__ phRe,
                            const float* __restrict__ phIm,
                            float* __restrict__ out,
                            int E, int N, int K) {
    int idx = blockIdx.x * blockDim.x + threadIdx.x;
    if (idx >= E * NORB2) return;
    int ab = idx % NORB2;
    int e  = idx / NORB2;
    int a  = ab / NORB, b = ab % NORB;

    float f = Ffac(a, b);
    if (f == 0.0f) return;                       // below-block-diagonal: no work
    float hf = hop[(long long)e * NORB2 + ab] * f;

    int ie = eidx[e];
    int je = eidx[E + e];

    long long L  = (long long)N * NORB;
    long long r1 = (long long)ie * NORB + a, c1 = (long long)je * NORB + b;
    long long r2 = (long long)je * NORB + b, c2 = (long long)ie * NORB + a;

    for (int k = 0; k < K; ++k) {
        float pr = phRe[(long long)k * E + e];
        float pi = phIm[(long long)k * E + e];
        long long base = (long long)k * L * L;
        long long p1 = (base + r1 * L + c1) * 2;
        long long p2 = (base + r2 * L + c2) * 2;
        atomicAdd(&out[p1],     pr * hf);
        atomicAdd(&out[p1 + 1], pi * hf);
        atomicAdd(&out[p2],     pr * hf);
        atomicAdd(&out[p2 + 1], -pi * hf);
    }
}

// ---------------------------------------------------------------------------
extern "C" void kernel_launch(void* const* d_in, const int* in_sizes, int n_in,
                              void* d_out, int out_size, void* d_ws, size_t ws_size,
                              hipStream_t stream) {
    const float* hopping = (const float*)d_in[0];   // [E, 169]
    const float* onsite  = (const float*)d_in[1];   // [N, 169]
    const float* kpoints = (const float*)d_in[2];   // [K, 3]
    const float* cshift  = (const float*)d_in[3];   // [E, 3]
    const int*   eidx    = (const int*)d_in[4];     // [2, E]
    float* out = (float*)d_out;                     // complex64 -> interleaved f32

    const int E = in_sizes[0] / NORB2;              // 2048
    const int N = in_sizes[1] / NORB2;              // 128
    const int K = in_sizes[2] / 3;                  // 16

    float* phRe = (float*)d_ws;                     // [K, E]
    float* phIm = phRe + (size_t)K * E;             // [K, E]  (256 KB total)

    // 1) zero output
    {
        long long n  = (long long)out_size;
        long long n4 = (n + 3) / 4;
        int blocks = (int)((n4 + 255) / 256);
        zero_kernel<<<blocks, 256, 0, stream>>>(out, n);
    }
    // 2) phases: WMMA path (one wave per 16-edge tile) when K==16 & E%16==0
    if (K == 16 && (E % 16) == 0) {
        phase_wmma_kernel<<<E / 16, 32, 0, stream>>>(kpoints, cshift, phRe, phIm, E);
    } else {
        int total = K * E;
        phase_scalar_kernel<<<(total + 255) / 256, 256, 0, stream>>>(
            kpoints, cshift, phRe, phIm, E, K);
    }
    // 3) onsite diagonal blocks
    {
        int total = K * N * NORB2;
        onsite_kernel<<<(total + 255) / 256, 256, 0, stream>>>(onsite, out, N, K);
    }
    // 4) edge scatter with Hermitian mirror
    {
        int total = E * NORB2;
        edge_kernel<<<(total + 255) / 256, 256, 0, stream>>>(
            hopping, eidx, phRe, phIm, out, E, N, K);
    }
}